// StructureLayer_24077586661957
// MI455X (gfx1250) — compile-verified
//
#include <hip/hip_runtime.h>
#include <math.h>

#define H 128

typedef __attribute__((ext_vector_type(2))) float v2f;
typedef __attribute__((ext_vector_type(8))) float v8f;

// Order-preserving monotonic encoding of float -> uint so we can use
// integer atomicMax for the segment max (avoids relying on float atomicMax).
__device__ __forceinline__ unsigned ord_enc(float f) {
  unsigned u = __float_as_uint(f);
  return (u & 0x80000000u) ? ~u : (u | 0x80000000u);
}
__device__ __forceinline__ float ord_dec(unsigned u) {
  return (u & 0x80000000u) ? __uint_as_float(u & 0x7FFFFFFFu)
                           : __uint_as_float(~u);
}

// ---------------- pass 0: init scratch ----------------
__global__ void k_init(float* __restrict__ neigh, float* __restrict__ ssum,
                       unsigned* __restrict__ smax, int n_nodes) {
  long long i = (long long)blockIdx.x * blockDim.x + threadIdx.x;
  long long total = (long long)n_nodes * H;
  if (i < total) neigh[i] = 0.0f;
  if (i < n_nodes) { ssum[i] = 0.0f; smax[i] = 0u; }  // 0 < enc(any finite f32)
}

// ---------------- pass 1: edge score + segment max ----------------
// One wave32 per edge; each lane owns 4 contiguous features (float4 = b128 load).
__global__ void k_score(const float* __restrict__ ent, const float* __restrict__ rel,
                        const int* __restrict__ src, const int* __restrict__ dst,
                        const int* __restrict__ rid,
                        float* __restrict__ score, unsigned* __restrict__ smax,
                        int n_edge) {
  int lane = threadIdx.x & 31;
  int e = blockIdx.x * (blockDim.x >> 5) + (threadIdx.x >> 5);
  if (e >= n_edge) return;
  int s = src[e], d = dst[e], r = rid[e];
  const float4* es = (const float4*)(ent + (long long)s * H);
  const float4* er = (const float4*)(rel + (long long)r * H);
  const float4* ed = (const float4*)(ent + (long long)d * H);
  float4 a = es[lane], b = er[lane], c = ed[lane];
  float4 comp = {a.x + b.x, a.y + b.y, a.z + b.z, a.w + b.w};
  float p = comp.x * c.x + comp.y * c.y + comp.z * c.z + comp.w * c.w;
  #pragma unroll
  for (int m = 16; m >= 1; m >>= 1) p += __shfl_xor(p, m, 32);
  if (lane == 0) {
    score[e] = p;
    atomicMax(&smax[d], ord_enc(p));
  }
}

// ---------------- pass 2: exp + segment sum ----------------
__global__ void k_expsum(const int* __restrict__ dst, const unsigned* __restrict__ smax,
                         float* __restrict__ score, float* __restrict__ ssum,
                         int n_edge) {
  int e = blockIdx.x * blockDim.x + threadIdx.x;
  if (e >= n_edge) return;
  int d = dst[e];
  float ex = __expf(score[e] - ord_dec(smax[d]));
  score[e] = ex;  // reuse buffer: score now holds exp(score - max)
  atomicAdd(&ssum[d], ex);
}

// ---------------- pass 3: normalize + weighted scatter ----------------
// One wave32 per edge; 4 f32 atomics per lane into L2-resident neigh[N,128].
__global__ void k_scatter(const float* __restrict__ ent, const float* __restrict__ rel,
                          const int* __restrict__ src, const int* __restrict__ dst,
                          const int* __restrict__ rid,
                          const float* __restrict__ score, const float* __restrict__ ssum,
                          float* __restrict__ neigh, int n_edge) {
  int lane = threadIdx.x & 31;
  int e = blockIdx.x * (blockDim.x >> 5) + (threadIdx.x >> 5);
  if (e >= n_edge) return;
  int s = src[e], d = dst[e], r = rid[e];
  float norm = score[e] / ssum[d];
  const float4* es = (const float4*)(ent + (long long)s * H);
  const float4* er = (const float4*)(rel + (long long)r * H);
  float4 a = es[lane], b = er[lane];
  float* out = neigh + (long long)d * H + lane * 4;
  atomicAdd(out + 0, (a.x + b.x) * norm);
  atomicAdd(out + 1, (a.y + b.y) * norm);
  atomicAdd(out + 2, (a.z + b.z) * norm);
  atomicAdd(out + 3, (a.w + b.w) * norm);
}

// ---------------- pass 4: out = tanh(neigh @ W) via V_WMMA_F32_16X16X4_F32 ----
// One wave per 16x16 output tile, K=128 in 32 wmma steps (full fp32 precision).
// Layouts per CDNA5 ISA 7.12.2:
//   A 16x4 f32: lanes 0-15 hold {K=0,K=1} for M=lane; lanes 16-31 hold {K=2,K=3}.
//   B  4x16 f32: mirrored, N striped across lanes.
//   D: VGPR r -> row M = r + 8*(lane>=16), col N = lane&15.
__global__ void k_gemm_tanh(const float* __restrict__ neigh, const float* __restrict__ W,
                            float* __restrict__ out, int n_nodes) {
  int lane = threadIdx.x & 31;
  int tile = blockIdx.x * (blockDim.x >> 5) + (threadIdx.x >> 5);
  const int tilesPerRow = H / 16;  // 8
  int rowTile = tile / tilesPerRow;
  int colTile = tile - rowTile * tilesPerRow;
  int row0 = rowTile * 16;
  if (row0 >= n_nodes) return;       // wave-uniform: EXEC stays all-ones
  int col0 = colTile * 16;
  int half = lane >> 4;              // 0 or 1
  int l = lane & 15;

  int arow = row0 + l;
  if (arow >= n_nodes) arow = n_nodes - 1;  // safety (N%16==0 in practice)
  const float* aptr = neigh + (long long)arow * H;

  v8f acc = {};
  #pragma unroll
  for (int k0 = 0; k0 < H; k0 += 4) {
    int ka = k0 + 2 * half;
    v2f av, bv;
    av.x = aptr[ka];
    av.y = aptr[ka + 1];
    bv.x = W[(long long)ka * H + col0 + l];
    bv.y = W[(long long)(ka + 1) * H + col0 + l];
    acc = __builtin_amdgcn_wmma_f32_16x16x4_f32(
        /*neg_a=*/false, av, /*neg_b=*/false, bv,
        /*c_mod=*/(short)0, acc, /*reuse_a=*/false, /*reuse_b=*/false);
  }

  #pragma unroll
  for (int r = 0; r < 8; ++r) {
    int m = row0 + r + 8 * half;
    if (m < n_nodes) out[(long long)m * H + col0 + l] = tanhf(acc[r]);
  }
}

extern "C" void kernel_launch(void* const* d_in, const int* in_sizes, int n_in,
                              void* d_out, int out_size, void* d_ws, size_t ws_size,
                              hipStream_t stream) {
  const float* ent = (const float*)d_in[0];
  const float* rel = (const float*)d_in[1];
  const float* W   = (const float*)d_in[2];
  const int*   src = (const int*)d_in[3];
  const int*   dst = (const int*)d_in[4];
  const int*   rid = (const int*)d_in[5];
  const int n_nodes = in_sizes[0] / H;
  const int n_edge  = in_sizes[3];

  // Workspace carve-up (256B aligned regions).
  char* ws = (char*)d_ws;
  size_t off = 0;
  auto carve = [&](size_t bytes) -> void* {
    void* p = ws + off;
    off += (bytes + 255) & ~(size_t)255;
    return p;
  };
  float*    score = (float*)carve((size_t)n_edge * sizeof(float));
  unsigned* smax  = (unsigned*)carve((size_t)n_nodes * sizeof(unsigned));
  float*    ssum  = (float*)carve((size_t)n_nodes * sizeof(float));
  float*    neigh = (float*)carve((size_t)n_nodes * H * sizeof(float));
  (void)ws_size; (void)n_in; (void)out_size;

  {
    long long total = (long long)n_nodes * H;
    int threads = 256;
    int blocks = (int)((total + threads - 1) / threads);
    k_init<<<blocks, threads, 0, stream>>>(neigh, ssum, smax, n_nodes);
  }
  {
    int threads = 256, wpb = threads / 32;
    int blocks = (n_edge + wpb - 1) / wpb;
    k_score<<<blocks, threads, 0, stream>>>(ent, rel, src, dst, rid, score, smax, n_edge);
  }
  {
    int threads = 256;
    int blocks = (n_edge + threads - 1) / threads;
    k_expsum<<<blocks, threads, 0, stream>>>(dst, smax, score, ssum, n_edge);
  }
  {
    int threads = 256, wpb = threads / 32;
    int blocks = (n_edge + wpb - 1) / wpb;
    k_scatter<<<blocks, threads, 0, stream>>>(ent, rel, src, dst, rid, score, ssum, neigh, n_edge);
  }
  {
    int tiles = ((n_nodes + 15) / 16) * (H / 16);
    int threads = 256, wpb = threads / 32;
    int blocks = (tiles + wpb - 1) / wpb;
    k_gemm_tanh<<<blocks, threads, 0, stream>>>(neigh, W, (float*)d_out, n_nodes);
  }
}